// Barrier_Net_50354196578949
// MI455X (gfx1250) — compile-verified
//
#include <hip/hip_runtime.h>
#include <math.h>

typedef __attribute__((ext_vector_type(2))) float v2f;
typedef __attribute__((ext_vector_type(8))) float v8f;

#define NNB 16
#define NOB 32
#define XDIM 133

// ---- LDS layout (float offsets) ----
// padded weights (block-shared)
#define OFF_PNW1 0          // 4x64
#define OFF_PNW2 256        // 64x16
#define OFF_RNW1 1280       // 16x64
#define OFF_RNW2 2304       // 64x16 (cols 8..15 zero)
#define OFF_POW1 3328       // 4x64  (rows 2..3 zero)
#define OFF_POW2 3584       // 64x16
#define OFF_ROW1 4608       // 16x64
#define OFF_ROW2 5632       // 64x16 (cols 8..15 zero)
#define OFF_PSW1 6656       // 20x64 (rows 18..19 zero)
#define OFF_PSW2 7936       // 64x64
#define OFF_PSW3 12032      // 64x16 (cols 2..15 zero)
#define OFF_PNB1 13056
#define OFF_PNB2 13120
#define OFF_RNB1 13136
#define OFF_RNB2 13200      // 16 (8..15 zero)
#define OFF_POB1 13216
#define OFF_POB2 13280
#define OFF_ROB1 13296
#define OFF_ROB2 13360      // 16 (8..15 zero)
#define OFF_PSB1 13376
#define OFF_PSB2 13440
#define OFF_PSB3 13504      // 16 (2..15 zero)
#define W_TOTAL  13520

// per-wave scratch (float offsets inside wave region)
#define S64 68              // padded stride for 64-wide buffers (bank-conflict free)
#define S20 20              // stride for 20-wide buffers
#define PW_A    0                   // 16*S64 : pooled_n / u_n / ps hidden1
#define PW_B    (PW_A + 16*S64)     // 16*S64 : pooled_o / u_o / ps hidden2
#define PW_T    (PW_B + 16*S64)     // 16*S20 : 16-dim intermediates
#define PW_H    (PW_T + 16*S20)     // 16*S20 : deepsets features [rho_n|rho_o|g|pad]
#define PW_EMP  (PW_H + 16*S20)     // 16*2   : "empty" controller output
#define PW_SIZE (PW_EMP + 32)
#define LDS_TOTAL (W_TOTAL + 4*PW_SIZE)

__device__ __forceinline__ v8f wmma4(v2f a, v2f b, v8f c) {
  // D = A(16x4,f32) * B(4x16,f32) + C(16x16,f32)
  return __builtin_amdgcn_wmma_f32_16x16x4_f32(false, a, false, b, (short)0, c,
                                               false, false);
}

// cross-half (lane <-> lane^16) add via v_permlanex16_b32: pure VALU, no LDS.
// identity nibble selects: lane i of each 16-group reads lane i of the other group.
__device__ __forceinline__ float halfswap_add(float s) {
  int si = __builtin_bit_cast(int, s);
  int o = __builtin_amdgcn_permlanex16(si, si, 0x76543210, 0xfedcba98, true, false);
  return s + __builtin_bit_cast(float, o);
}

// B-fragment: row-major W (ncols columns), rows k0..k0+3, cols col0..col0+15
__device__ __forceinline__ v2f bfrag(const float* W, int ncols, int k0, int col0,
                                     int lane) {
  int r = k0 + 2 * (lane >> 4);
  int c = col0 + (lane & 15);
  v2f b;
  b.x = W[r * ncols + c];
  b.y = W[(r + 1) * ncols + c];
  return b;
}

// A-fragment: row-major buf (stride), rows = M(16), cols k0..k0+3
__device__ __forceinline__ v2f afrag(const float* buf, int stride, int k0, int lane) {
  int m = lane & 15;
  int k = k0 + 2 * (lane >> 4);
  v2f a;
  a.x = buf[m * stride + k];
  a.y = buf[m * stride + k + 1];
  return a;
}

// C = A(16xK) @ W(K x ncols)[:, col0:col0+16], K multiple-of-4 via 16x16x4 chain
__device__ __forceinline__ v8f gemm_tile(const float* A, int astride, const float* W,
                                         int ncols, int K, int col0, int lane) {
  v8f c = {};
  for (int k = 0; k < K; k += 4) {
    v2f a = afrag(A, astride, k, lane);
    v2f b = bfrag(W, ncols, k, col0, lane);
    c = wmma4(a, b, c);
  }
  return c;
}

// C-tile (M=element, N=feature) -> row-major buf
__device__ __forceinline__ void store_tile(float* buf, int stride, int col0, v8f c,
                                           int lane) {
  int hl = lane >> 4;
  int col = col0 + (lane & 15);
#pragma unroll
  for (int v = 0; v < 8; ++v) buf[(v + 8 * hl) * stride + col] = c[v];
}

extern "C" __global__ __launch_bounds__(128)
void barrier_net_kernel(const float* __restrict__ x, const float* __restrict__ noise,
                        const float* pnW1, const float* pnb1, const float* pnW2,
                        const float* pnb2, const float* rnW1, const float* rnb1,
                        const float* rnW2, const float* rnb2, const float* poW1,
                        const float* pob1, const float* poW2, const float* pob2,
                        const float* roW1, const float* rob1, const float* roW2,
                        const float* rob2, const float* psW1, const float* psb1,
                        const float* psW2, const float* psb2, const float* psW3,
                        const float* psb3, float* __restrict__ out) {
  extern __shared__ float smem[];
  const int tid = threadIdx.x;
  const int wave = tid >> 5;
  const int lane = tid & 31;
  const int hl = lane >> 4;
  const int lm = lane & 15;

  // ---- stage weights into LDS (zero-fill padding first) ----
  for (int i = tid; i < W_TOTAL; i += 128) smem[i] = 0.f;
  __syncthreads();
  for (int i = tid; i < 4 * 64; i += 128) smem[OFF_PNW1 + i] = pnW1[i];
  for (int i = tid; i < 64 * 16; i += 128) smem[OFF_PNW2 + i] = pnW2[i];
  for (int i = tid; i < 16 * 64; i += 128) smem[OFF_RNW1 + i] = rnW1[i];
  for (int i = tid; i < 64 * 8; i += 128) smem[OFF_RNW2 + (i >> 3) * 16 + (i & 7)] = rnW2[i];
  for (int i = tid; i < 2 * 64; i += 128) smem[OFF_POW1 + i] = poW1[i];
  for (int i = tid; i < 64 * 16; i += 128) smem[OFF_POW2 + i] = poW2[i];
  for (int i = tid; i < 16 * 64; i += 128) smem[OFF_ROW1 + i] = roW1[i];
  for (int i = tid; i < 64 * 8; i += 128) smem[OFF_ROW2 + (i >> 3) * 16 + (i & 7)] = roW2[i];
  for (int i = tid; i < 18 * 64; i += 128) smem[OFF_PSW1 + i] = psW1[i];
  for (int i = tid; i < 64 * 64; i += 128) smem[OFF_PSW2 + i] = psW2[i];
  for (int i = tid; i < 64 * 2; i += 128) smem[OFF_PSW3 + (i >> 1) * 16 + (i & 1)] = psW3[i];
  for (int i = tid; i < 64; i += 128) {
    smem[OFF_PNB1 + i] = pnb1[i];
    smem[OFF_RNB1 + i] = rnb1[i];
    smem[OFF_POB1 + i] = pob1[i];
    smem[OFF_ROB1 + i] = rob1[i];
    smem[OFF_PSB1 + i] = psb1[i];
    smem[OFF_PSB2 + i] = psb2[i];
  }
  for (int i = tid; i < 16; i += 128) {
    smem[OFF_PNB2 + i] = pnb2[i];
    smem[OFF_POB2 + i] = pob2[i];
  }
  for (int i = tid; i < 8; i += 128) {
    smem[OFF_RNB2 + i] = rnb2[i];
    smem[OFF_ROB2 + i] = rob2[i];
  }
  for (int i = tid; i < 2; i += 128) smem[OFF_PSB3 + i] = psb3[i];
  __syncthreads();

  const float* W = smem;
  float* pw = smem + W_TOTAL + wave * PW_SIZE;
  const int base = blockIdx.x * 64 + wave * 16;

  // ---- hoist loop-invariant L1 B-fragments + biases into registers ----
  v2f bn[4], bo[4];
  float biasn[4], biaso[4];
#pragma unroll
  for (int j = 0; j < 4; ++j) {
    bn[j] = bfrag(W + OFF_PNW1, 64, 0, j * 16, lane);
    bo[j] = bfrag(W + OFF_POW1, 64, 0, j * 16, lane);
    biasn[j] = W[OFF_PNB1 + j * 16 + lm];
    biaso[j] = W[OFF_POB1 + j * 16 + lm];
  }

  // ---- stage 1: pooled_n[e] = sum_i relu(neigh_i @ pnW1 + pnb1)  (DeepSets pool)
  for (int e = 0; e < 16; ++e) {
    const float* xb = x + (size_t)(base + e) * XDIM;
    v2f a;  // A = 16 neighbors x 4 feats (lanes 0-15: K0,K1; lanes 16-31: K2,K3)
    a.x = xb[5 + lm * 4 + hl * 2];
    a.y = xb[5 + lm * 4 + hl * 2 + 1];
#pragma unroll
    for (int j = 0; j < 4; ++j) {
      v8f c = {};
      c = wmma4(a, bn[j], c);
      float s = 0.f;
#pragma unroll
      for (int v = 0; v < 8; ++v) s += fmaxf(c[v] + biasn[j], 0.f);
      s = halfswap_add(s);  // both halves now hold the full column sum
      // lanes L and L+16 write the identical value to the same slot (benign)
      pw[PW_A + e * S64 + j * 16 + lm] = s;
    }
  }

  // ---- stage 2: pooled_o[e] = sum over 32 obstacles (2 M-tiles, K=2 padded to 4)
  for (int e = 0; e < 16; ++e) {
    const float* xb = x + (size_t)(base + e) * XDIM;
    v2f a0, a1;  // lanes 16-31 carry K=2,3 which are zero-padded
    a0.x = (hl == 0) ? xb[69 + lm * 2] : 0.f;
    a0.y = (hl == 0) ? xb[69 + lm * 2 + 1] : 0.f;
    a1.x = (hl == 0) ? xb[69 + (16 + lm) * 2] : 0.f;
    a1.y = (hl == 0) ? xb[69 + (16 + lm) * 2 + 1] : 0.f;
#pragma unroll
    for (int j = 0; j < 4; ++j) {
      v8f c0 = {}, c1 = {};
      c0 = wmma4(a0, bo[j], c0);
      c1 = wmma4(a1, bo[j], c1);
      float s = 0.f;
#pragma unroll
      for (int v = 0; v < 8; ++v) {
        s += fmaxf(c0[v] + biaso[j], 0.f);
        s += fmaxf(c1[v] + biaso[j], 0.f);
      }
      s = halfswap_add(s);
      pw[PW_B + e * S64 + j * 16 + lm] = s;
    }
  }

  // ---- neighbor rho chain: t_n = pooled_n @ pnW2 + 16*pnb2
  {
    v8f c = gemm_tile(pw + PW_A, S64, W + OFF_PNW2, 16, 64, 0, lane);
    float bias = 16.f * W[OFF_PNB2 + lm];
#pragma unroll
    for (int v = 0; v < 8; ++v) c[v] += bias;
    store_tile(pw + PW_T, S20, 0, c, lane);
  }
  // u_n = relu(t_n @ rnW1 + rnb1) -> bufA (pooled_n no longer needed)
  for (int j = 0; j < 4; ++j) {
    v8f c = gemm_tile(pw + PW_T, S20, W + OFF_RNW1, 64, 16, j * 16, lane);
    float bias = W[OFF_RNB1 + j * 16 + lm];
#pragma unroll
    for (int v = 0; v < 8; ++v) c[v] = fmaxf(c[v] + bias, 0.f);
    store_tile(pw + PW_A, S64, j * 16, c, lane);
  }
  // rho_n = u_n @ rnW2(pad) + rnb2 -> H[:, 0..7]
  {
    v8f c = gemm_tile(pw + PW_A, S64, W + OFF_RNW2, 16, 64, 0, lane);
    float bias = W[OFF_RNB2 + lm];
    if (lm < 8) {
#pragma unroll
      for (int v = 0; v < 8; ++v) pw[PW_H + (v + 8 * hl) * S20 + lm] = c[v] + bias;
    }
  }

  // ---- obstacle rho chain: t_o = pooled_o @ poW2 + 32*pob2
  {
    v8f c = gemm_tile(pw + PW_B, S64, W + OFF_POW2, 16, 64, 0, lane);
    float bias = 32.f * W[OFF_POB2 + lm];
#pragma unroll
    for (int v = 0; v < 8; ++v) c[v] += bias;
    store_tile(pw + PW_T, S20, 0, c, lane);
  }
  for (int j = 0; j < 4; ++j) {
    v8f c = gemm_tile(pw + PW_T, S20, W + OFF_ROW1, 64, 16, j * 16, lane);
    float bias = W[OFF_ROB1 + j * 16 + lm];
#pragma unroll
    for (int v = 0; v < 8; ++v) c[v] = fmaxf(c[v] + bias, 0.f);
    store_tile(pw + PW_B, S64, j * 16, c, lane);
  }
  {
    v8f c = gemm_tile(pw + PW_B, S64, W + OFF_ROW2, 16, 64, 0, lane);
    float bias = W[OFF_ROB2 + lm];
    if (lm < 8) {
#pragma unroll
      for (int v = 0; v < 8; ++v) pw[PW_H + (v + 8 * hl) * S20 + 8 + lm] = c[v] + bias;
    }
  }

  // ---- goal features + padding into H[:, 16..19] (both halves write same value)
  {
    const float* xg = x + (size_t)(base + lm) * XDIM;
    pw[PW_H + lm * S20 + 16] = xg[1];
    pw[PW_H + lm * S20 + 17] = xg[2];
    pw[PW_H + lm * S20 + 18] = 0.f;
    pw[PW_H + lm * S20 + 19] = 0.f;
  }

  // ---- policy MLP: h1 = relu(H @ psW1 + psb1) (K=20 padded)
  for (int j = 0; j < 4; ++j) {
    v8f c = gemm_tile(pw + PW_H, S20, W + OFF_PSW1, 64, 20, j * 16, lane);
    float bias = W[OFF_PSB1 + j * 16 + lm];
#pragma unroll
    for (int v = 0; v < 8; ++v) c[v] = fmaxf(c[v] + bias, 0.f);
    store_tile(pw + PW_A, S64, j * 16, c, lane);
  }
  // h2 = relu(h1 @ psW2 + psb2)
  for (int j = 0; j < 4; ++j) {
    v8f c = gemm_tile(pw + PW_A, S64, W + OFF_PSW2, 64, 64, j * 16, lane);
    float bias = W[OFF_PSB2 + j * 16 + lm];
#pragma unroll
    for (int v = 0; v < 8; ++v) c[v] = fmaxf(c[v] + bias, 0.f);
    store_tile(pw + PW_B, S64, j * 16, c, lane);
  }
  // empty = 2*tanh(h2 @ psW3 + psb3)   ((t+1)/2*(PHI_MAX-PHI_MIN)+PHI_MIN == 2t)
  {
    v8f c = gemm_tile(pw + PW_B, S64, W + OFF_PSW3, 16, 64, 0, lane);
    if (lm < 2) {
      float bias = W[OFF_PSB3 + lm];
#pragma unroll
      for (int v = 0; v < 8; ++v)
        pw[PW_EMP + (v + 8 * hl) * 2 + lm] = 2.f * tanhf(c[v] + bias);
    }
  }

  // ---- epilogue: CBF barrier term + tanh squash. 2 lanes per element.
  {
    const int e = lane >> 1;
    const int d = lane & 1;
    const int bidx = base + e;
    const float* xb = x + (size_t)bidx * XDIM;
    float bar = 0.f;
#pragma unroll 4
    for (int i = 0; i < NNB; ++i) {
      float px = xb[5 + i * 4], py = xb[5 + i * 4 + 1];
      float nrm = sqrtf(px * px + py * py);
      float pd = (d == 0) ? px : py;
      bar += 0.01f * (-pd) / (nrm * (nrm - 0.3f));
    }
#pragma unroll 4
    for (int i = 0; i < NOB; ++i) {
      float px = xb[69 + i * 2], py = xb[69 + i * 2 + 1];
      float nrm = sqrtf(px * px + py * py);
      float pd = (d == 0) ? px : py;
      bar += 0.01f * (-pd) / (nrm * (nrm - 0.5f));
    }
    float z = pw[PW_EMP + e * 2 + d] + bar + noise[(size_t)bidx * 2 + d];
    out[(size_t)bidx * 2 + d] = 2.f * tanhf(z);
  }
}

extern "C" void kernel_launch(void* const* d_in, const int* in_sizes, int n_in,
                              void* d_out, int out_size, void* d_ws, size_t ws_size,
                              hipStream_t stream) {
  const float* x = (const float*)d_in[0];
  const float* noise = (const float*)d_in[1];
  const float* pnW1 = (const float*)d_in[2];
  const float* pnb1 = (const float*)d_in[3];
  const float* pnW2 = (const float*)d_in[4];
  const float* pnb2 = (const float*)d_in[5];
  const float* rnW1 = (const float*)d_in[6];
  const float* rnb1 = (const float*)d_in[7];
  const float* rnW2 = (const float*)d_in[8];
  const float* rnb2 = (const float*)d_in[9];
  const float* poW1 = (const float*)d_in[10];
  const float* pob1 = (const float*)d_in[11];
  const float* poW2 = (const float*)d_in[12];
  const float* pob2 = (const float*)d_in[13];
  const float* roW1 = (const float*)d_in[14];
  const float* rob1 = (const float*)d_in[15];
  const float* roW2 = (const float*)d_in[16];
  const float* rob2 = (const float*)d_in[17];
  const float* psW1 = (const float*)d_in[18];
  const float* psb1 = (const float*)d_in[19];
  const float* psW2 = (const float*)d_in[20];
  const float* psb2 = (const float*)d_in[21];
  const float* psW3 = (const float*)d_in[22];
  const float* psb3 = (const float*)d_in[23];
  float* out = (float*)d_out;

  const int Bn = in_sizes[0] / XDIM;       // 65536
  dim3 grid(Bn / 64), block(128);          // 4 waves x 16 elements per block
  size_t shmem = (size_t)LDS_TOTAL * sizeof(float);
  barrier_net_kernel<<<grid, block, shmem, stream>>>(
      x, noise, pnW1, pnb1, pnW2, pnb2, rnW1, rnb1, rnW2, rnb2, poW1, pob1, poW2,
      pob2, roW1, rob1, roW2, rob2, psW1, psb1, psW2, psb2, psW3, psb3, out);
}